// RKHSRendererGlobalScale_33792802685021
// MI455X (gfx1250) — compile-verified
//
#include <hip/hip_runtime.h>
#include <hip/hip_bf16.h>
#include <math.h>

#define HH 256
#define WW 256
#define TILE_SZ 64
#define NG 1024
#define NFIELD 12

typedef __attribute__((ext_vector_type(2))) float v2f;
typedef __attribute__((ext_vector_type(8))) float v8f;

// ---------------------------------------------------------------------------
// Kernel 1: per-Gaussian preprocessing (projection, rects, -0.5/sigma^2)
// Writes 12 SoA fields of length NG into ws[0 .. 12*NG)
//   0:depth 1:m2x 2:m2y 3:s2i(=-0.5/s2sq) 4:op 5:cr 6:cg 7:cb
//   8:rminx 9:rminy 10:rmaxx 11:rmaxy
// ---------------------------------------------------------------------------
__global__ void rgs_preprocess(const float* __restrict__ means3d,
                               const float* __restrict__ opacity,
                               const float* __restrict__ scale3d,
                               const float* __restrict__ features,
                               const float* __restrict__ viewm,
                               const float* __restrict__ projm,
                               float* __restrict__ ws) {
  int i = blockIdx.x * blockDim.x + threadIdx.x;
  if (i >= NG) return;
  float x = means3d[3*i+0], y = means3d[3*i+1], z = means3d[3*i+2];
  float pv[4], ph[4];
#pragma unroll
  for (int j = 0; j < 4; ++j)
    pv[j] = x*viewm[0*4+j] + y*viewm[1*4+j] + z*viewm[2*4+j] + viewm[3*4+j];
#pragma unroll
  for (int j = 0; j < 4; ++j)
    ph[j] = pv[0]*projm[0*4+j] + pv[1]*projm[1*4+j] + pv[2]*projm[2*4+j] + pv[3]*projm[3*4+j];
  float invw = 1.0f / (ph[3] + 1e-6f);
  float ppx = ph[0]*invw, ppy = ph[1]*invw;
  float depth = pv[2];
  float tz = pv[2];                               // t == pv[0..2] (row-vector convention)
  const float FOCAL = (float)WW / (2.0f * 0.5f);  // W / (2*TAN_FOV)
  float s2  = scale3d[i] * FOCAL / tz;            // scale2d (both comps equal)
  float mx  = ((ppx + 1.0f)*(float)WW - 1.0f)*0.5f;
  float my  = ((ppy + 1.0f)*(float)HH - 1.0f)*0.5f;
  float rad = s2 * 5.0f;                          // RADII_MULT
  float s2i = -0.5f / (s2*s2);
  float rminx = fminf(fmaxf(mx - rad, 0.0f), (float)(WW-1));
  float rminy = fminf(fmaxf(my - rad, 0.0f), (float)(HH-1));
  float rmaxx = fminf(fmaxf(mx + rad, 0.0f), (float)(WW-1));
  float rmaxy = fminf(fmaxf(my + rad, 0.0f), (float)(HH-1));
  ws[0*NG+i]  = depth;
  ws[1*NG+i]  = mx;
  ws[2*NG+i]  = my;
  ws[3*NG+i]  = s2i;
  ws[4*NG+i]  = opacity[i];
  ws[5*NG+i]  = features[3*i+0];
  ws[6*NG+i]  = features[3*i+1];
  ws[7*NG+i]  = features[3*i+2];
  ws[8*NG+i]  = rminx;
  ws[9*NG+i]  = rminy;
  ws[10*NG+i] = rmaxx;
  ws[11*NG+i] = rmaxy;
}

// ---------------------------------------------------------------------------
// Kernel 2: stable O(N^2) rank sort by depth; scatter all fields into
// ws[12*NG .. 24*NG) in front-to-back order.
// ---------------------------------------------------------------------------
__global__ void rgs_sort_scatter(float* __restrict__ ws) {
  int i = blockIdx.x * blockDim.x + threadIdx.x;
  if (i >= NG) return;
  const float* depth = ws;
  float di = depth[i];
  int rank = 0;
  for (int j = 0; j < NG; ++j) {
    float dj = depth[j];
    rank += (dj < di) || (dj == di && j < i);   // stable tie-break == argsort
  }
  float* s = ws + NFIELD*NG;
#pragma unroll
  for (int f = 0; f < NFIELD; ++f)
    s[f*NG + rank] = ws[f*NG + i];
}

// ---------------------------------------------------------------------------
// Kernel 3: render. One wave == 16 pixels. Superchunks of 32 sorted
// Gaussians: each lane stages one distinct Gaussian; two
// V_WMMA_F32_16X16X4_F32 ops compute the two 16x16 dx2 tiles; ordered
// alpha compositing with half-wave affine merge via shfl_xor(16).
// Grid: 16 tiles * 32 segs = 512 blocks of 256 threads (8 waves).
// ---------------------------------------------------------------------------
__global__ void __launch_bounds__(256)
rgs_render(const float* __restrict__ ws, float* __restrict__ out) {
  const float* s    = ws + NFIELD*NG;
  const float* sdep = s + 0*NG;
  const float* sm2x = s + 1*NG;
  const float* sm2y = s + 2*NG;
  const float* ss2i = s + 3*NG;
  const float* sop  = s + 4*NG;
  const float* scr  = s + 5*NG;
  const float* scg  = s + 6*NG;
  const float* scb  = s + 7*NG;
  const float* srnx = s + 8*NG;
  const float* srny = s + 9*NG;
  const float* srxx = s + 10*NG;
  const float* srxy = s + 11*NG;

  int lane = threadIdx.x & 31;
  int wave = threadIdx.x >> 5;
  int tile = blockIdx.x >> 5;     // 0..15
  int seg  = blockIdx.x & 31;     // 0..31
  int grp  = seg*8 + wave;        // 0..255 : 16-pixel strip within the 64x64 tile
  int ly   = grp >> 2;            // row 0..63
  int lx   = (grp & 3) << 4;      // col start 0/16/32/48
  int tu   = tile & 3, tv = tile >> 2;
  float u0 = (float)(tu * TILE_SZ);
  float v0 = (float)(tv * TILE_SZ);
  int  pxi = tu*TILE_SZ + lx + (lane & 15);
  int  pyi = tv*TILE_SZ + ly;
  float px = (float)pxi, py = (float)pyi;
  bool hi  = lane >= 16;

  // B fragment (4x16 f32, pixels in columns). VGPR0 = K{0|2}, VGPR1 = K{1|3}.
  // K-vector per pixel: [1, px, py, px^2+py^2]
  v2f Bf;
  Bf.x = hi ? py : 1.0f;
  Bf.y = hi ? (px*px + py*py) : px;

  float Trun = 1.0f;                     // persistent per-pixel transmittance
  float accr = 0.f, accg = 0.f, accb = 0.f, accd = 0.f, acca = 0.f;

  for (int sc = 0; sc < NG/32; ++sc) {
    int gl = sc*32 + lane;               // each lane stages a DISTINCT Gaussian
    int pf = (gl + 32) & (NG - 1);       // next superchunk (wrapped, in-bounds)
    __builtin_prefetch(&ss2i[pf], 0, 3); // -> global_prefetch_b8
    __builtin_prefetch(&scr[pf],  0, 3);

    float mx  = sm2x[gl], my = sm2y[gl];
    float s2i = ss2i[gl], dpv = sdep[gl];
    float crv = scr[gl],  cgv = scg[gl], cbv = scb[gl];
    float tlx = fmaxf(srnx[gl], u0);
    float tly = fmaxf(srny[gl], v0);
    float brx = fminf(srxx[gl], u0 + (float)(TILE_SZ-1));
    float bry = fminf(srxy[gl], v0 + (float)(TILE_SZ-1));
    float mskv = ((brx > tlx) && (bry > tly)) ? 1.0f : 0.0f;
    // mask folded into opacity: min(gw*op,0.99)*mask == min(gw*(op*mask),0.99)
    float opm = sop[gl] * mskv;

    unsigned live = __builtin_amdgcn_ballot_w32(opm > 0.0f);
    if (live == 0u) continue;            // uniform: whole superchunk dead

    // Composite one 16-Gaussian chunk whose params are staged in lanes
    // [srcoff .. srcoff+15]. Front-to-back within chunk; merges into
    // (accr..acca, Trun).
    auto do_chunk16 = [&](int srcoff) {
      // Rebuild A-fragment: Gaussian (lane&15) of this chunk.
      float amx = __shfl(mx, srcoff + (lane & 15), 32);
      float amy = __shfl(my, srcoff + (lane & 15), 32);
      // K-vector per Gaussian: [mx^2+my^2, -2mx, -2my, 1]
      v2f Af;
      Af.x = hi ? (-2.0f*amy) : (amx*amx + amy*amy);
      Af.y = hi ? 1.0f        : (-2.0f*amx);
      v8f Cd = {};
      Cd = __builtin_amdgcn_wmma_f32_16x16x4_f32(
          /*neg_a=*/false, Af, /*neg_b=*/false, Bf,
          /*c_mod=*/(short)0, Cd, /*reuse_a=*/false, /*reuse_b=*/false);
      // D layout: lane holds pixel (lane%16); VGPR r = Gaussian r + 8*hi.

      float Tl = 1.f, sr = 0.f, sg = 0.f, sb = 0.f, sd = 0.f, sa = 0.f;
      int sbase = srcoff + (hi ? 8 : 0);
#pragma unroll
      for (int r = 0; r < 8; ++r) {
        int src = sbase + r;             // broadcast params from staging lane
        float m_s2i = __shfl(s2i, src, 32);
        float m_opm = __shfl(opm, src, 32);
        float m_dep = __shfl(dpv, src, 32);
        float m_cr  = __shfl(crv, src, 32);
        float m_cg  = __shfl(cgv, src, 32);
        float m_cb  = __shfl(cbv, src, 32);
        float gw    = __expf(Cd[r] * m_s2i);
        float alpha = fminf(gw * m_opm, 0.99f);
        float wgt   = alpha * Tl;
        sr += wgt*m_cr; sg += wgt*m_cg; sb += wgt*m_cb;
        sd += wgt*m_dep; sa += wgt;
        Tl *= (1.0f - alpha);
      }
      // Merge halves (lower half = earlier in depth order).
      float o_sr = __shfl_xor(sr, 16, 32);
      float o_sg = __shfl_xor(sg, 16, 32);
      float o_sb = __shfl_xor(sb, 16, 32);
      float o_sd = __shfl_xor(sd, 16, 32);
      float o_sa = __shfl_xor(sa, 16, 32);
      float o_T  = __shfl_xor(Tl, 16, 32);
      float Sr = hi ? (o_sr + o_T*sr) : (sr + Tl*o_sr);
      float Sg = hi ? (o_sg + o_T*sg) : (sg + Tl*o_sg);
      float Sb = hi ? (o_sb + o_T*sb) : (sb + Tl*o_sb);
      float Sd = hi ? (o_sd + o_T*sd) : (sd + Tl*o_sd);
      float Sa = hi ? (o_sa + o_T*sa) : (sa + Tl*o_sa);
      accr += Trun*Sr; accg += Trun*Sg; accb += Trun*Sb;
      accd += Trun*Sd; acca += Trun*Sa;
      Trun *= Tl * o_T;
    };

    if (live & 0xFFFFu) do_chunk16(0);   // Gaussians sc*32 + [0..15]
    if (live >> 16)     do_chunk16(16);  // Gaussians sc*32 + [16..31]
  }

  if (!hi) {                             // both halves hold identical state
    int pid = pyi*WW + pxi;
    float omA = 1.0f - acca;
    out[3*pid+0] = fminf(fmaxf(accr + omA, 0.0f), 1.0f);
    out[3*pid+1] = fminf(fmaxf(accg + omA, 0.0f), 1.0f);
    out[3*pid+2] = fminf(fmaxf(accb + omA, 0.0f), 1.0f);
    out[HH*WW*3 + pid] = accd;           // depths plane
    out[HH*WW*4 + pid] = acca;           // accs plane
  }
}

// ---------------------------------------------------------------------------
extern "C" void kernel_launch(void* const* d_in, const int* in_sizes, int n_in,
                              void* d_out, int out_size, void* d_ws, size_t ws_size,
                              hipStream_t stream) {
  (void)in_sizes; (void)n_in; (void)out_size; (void)ws_size;
  const float* means3d  = (const float*)d_in[0];
  const float* opacity  = (const float*)d_in[1];
  const float* scale3d  = (const float*)d_in[2];
  const float* features = (const float*)d_in[3];
  const float* viewm    = (const float*)d_in[4];
  const float* projm    = (const float*)d_in[5];
  float* ws  = (float*)d_ws;   // needs 2*12*1024 floats = 96 KB
  float* out = (float*)d_out;

  rgs_preprocess<<<(NG+255)/256, 256, 0, stream>>>(means3d, opacity, scale3d,
                                                   features, viewm, projm, ws);
  rgs_sort_scatter<<<(NG+255)/256, 256, 0, stream>>>(ws);
  rgs_render<<<16*32, 256, 0, stream>>>(ws, out);
}